// VisionMamba_8177617731790
// MI455X (gfx1250) — compile-verified
//
#include <hip/hip_runtime.h>
#include <hip/hip_bf16.h>
#include <math.h>

// ---------------------------------------------------------------------------
// Vision Mamba forward for MI455X (gfx1250), fp32 via V_WMMA_F32_16X16X4_F32.
// Each wave computes a 16x64 output tile (4 accumulators) so one A fragment
// feeds 4 WMMAs; wave-uniform fast path avoids the N-mask multiply entirely.
// ---------------------------------------------------------------------------

typedef __attribute__((ext_vector_type(2))) float v2f;
typedef __attribute__((ext_vector_type(8))) float v8f;

#define BATCH     64
#define LSEQ      196
#define D_MODEL   192
#define D_INNER   384
#define D_STATE   16
#define D_CONV    4
#define DT_RANK   12
#define DEPTH     24
#define NCLS      1000
#define IMG       224
#define PATCH     16
#define GRID      14
#define ROWS_BL   (BATCH * LSEQ)     // 12544, multiple of 16

__device__ __forceinline__ float silu_f(float x) {
    return x / (1.0f + __expf(-x));
}

__device__ __forceinline__ v8f wmma4(v2f a, v2f b, v8f c) {
    return __builtin_amdgcn_wmma_f32_16x16x4_f32(
        false, a, false, b, (short)0, c, false, false);
}

// ---------------------------------------------------------------------------
// Generic WMMA GEMM:  C[M,N] = A[M,K] * W[N,K]^T (+bias, optional softplus)
// One wave computes one 16x64 tile (4 x 16x16 subtiles sharing the A frag).
// Requires: M % 16 == 0, K % 4 == 0, lda/ldc even (8B-aligned float2 loads).
// ---------------------------------------------------------------------------
__global__ void gemm_wmma_f32(const float* __restrict__ A, long lda,
                              const float* __restrict__ W,     // (N, K) row-major
                              const float* __restrict__ bias,  // may be null
                              float* __restrict__ C, long ldc,
                              int M, int N, int K, int nblocks, int epilogue) {
    const int wave = threadIdx.x >> 5;
    const int lane = threadIdx.x & 31;
    const int tile = blockIdx.x * (blockDim.x >> 5) + wave;
    const int total = (M >> 4) * nblocks;
    if (tile >= total) return;                 // whole-wave uniform exit

    const int mt = tile / nblocks;
    const int nb = tile % nblocks;
    const int n0 = nb * 64;

    const int arow  = mt * 16 + (lane & 15);
    const int khalf = (lane >> 4) << 1;        // 0 for lanes 0-15, 2 for 16-31
    const float* ap = A + (long)arow * lda + khalf;

    v8f acc[4] = {};

    if (n0 + 64 <= N) {
        // -------- fast path: whole 64-wide block in range, no masking ------
        const float* bp = W + (long)(n0 + (lane & 15)) * K + khalf;
        const long sstep = (long)16 * K;       // 16 weight rows per subtile
        for (int kk = 0; kk < K; kk += 4) {
            float2 av = *(const float2*)(ap + kk);
            v2f a; a[0] = av.x; a[1] = av.y;
#pragma unroll
            for (int s = 0; s < 4; ++s) {
                float2 bv = *(const float2*)(bp + s * sstep + kk);
                v2f b; b[0] = bv.x; b[1] = bv.y;
                acc[s] = wmma4(a, b, acc[s]);
            }
        }
    } else {
        // -------- edge path: per-subtile branchless masking ----------------
        const float* bps[4];
        float msk[4];
#pragma unroll
        for (int s = 0; s < 4; ++s) {
            const int ncol = n0 + s * 16 + (lane & 15);
            const bool nv = (ncol < N);
            msk[s] = nv ? 1.0f : 0.0f;
            bps[s] = W + (long)(nv ? ncol : 0) * K + khalf;
        }
        for (int kk = 0; kk < K; kk += 4) {
            float2 av = *(const float2*)(ap + kk);
            v2f a; a[0] = av.x; a[1] = av.y;
#pragma unroll
            for (int s = 0; s < 4; ++s) {
                float2 bv = *(const float2*)(bps[s] + kk);
                v2f b; b[0] = bv.x * msk[s]; b[1] = bv.y * msk[s];
                acc[s] = wmma4(a, b, acc[s]);
            }
        }
    }

    // -------- epilogue + store (C: VGPR r -> row r, second half +8) --------
    const int mrow0 = mt * 16 + ((lane >> 4) << 3);
#pragma unroll
    for (int s = 0; s < 4; ++s) {
        const int ncol = n0 + s * 16 + (lane & 15);
        if (ncol < N) {
            const float badd = bias ? bias[ncol] : 0.0f;
#pragma unroll
            for (int r = 0; r < 8; ++r) {
                float v = acc[s][r] + badd;
                if (epilogue == 1)             // softplus
                    v = (v > 20.0f) ? v : log1pf(__expf(v));
                C[(long)(mrow0 + r) * ldc + ncol] = v;
            }
        }
    }
}

// ---------------------------------------------------------------------------
// Patch embedding as a WMMA GEMM with gathered A (NCHW image):
//   tokens[b*196 + i*14 + j, e] = sum_{c,p,q} x[b,c,i*16+p,j*16+q] * pw[e,c,p,q] + pb[e]
// K = 768 = (c,p,q); consecutive K pairs are contiguous in x (q innermost).
// One wave computes a 16x64 tile; N = 192 = 3 blocks, always full.
// ---------------------------------------------------------------------------
__global__ void patch_embed_wmma(const float* __restrict__ x,
                                 const float* __restrict__ pw,
                                 const float* __restrict__ pb,
                                 float* __restrict__ tokens) {
    const int nblocks = D_MODEL / 64;          // 3
    const int wave = threadIdx.x >> 5;
    const int lane = threadIdx.x & 31;
    const int tile = blockIdx.x * (blockDim.x >> 5) + wave;
    if (tile >= (ROWS_BL / 16) * nblocks) return;

    const int mt = tile / nblocks;
    const int nb = tile % nblocks;
    const int n0 = nb * 64;

    const int m  = mt * 16 + (lane & 15);
    const int b  = m / LSEQ;
    const int t  = m % LSEQ;
    const int ti = t / GRID;
    const int tj = t % GRID;
    const int khalf = (lane >> 4) << 1;
    const float* bp = pw + (long)(n0 + (lane & 15)) * 768 + khalf;
    const long sstep = (long)16 * 768;

    v8f acc[4] = {};
    for (int c = 0; c < 3; ++c) {
        for (int p = 0; p < PATCH; ++p) {
            const float* arow =
                x + ((long)(b * 3 + c) * IMG + ti * PATCH + p) * IMG
                  + tj * PATCH + khalf;
            const float* brow = bp + (c * 256 + p * 16);
#pragma unroll
            for (int q = 0; q < 16; q += 4) {
                float2 av = *(const float2*)(arow + q);
                v2f a; a[0] = av.x; a[1] = av.y;
#pragma unroll
                for (int s = 0; s < 4; ++s) {
                    float2 bv = *(const float2*)(brow + s * sstep + q);
                    v2f bb; bb[0] = bv.x; bb[1] = bv.y;
                    acc[s] = wmma4(a, bb, acc[s]);
                }
            }
        }
    }

    const int mrow0 = mt * 16 + ((lane >> 4) << 3);
#pragma unroll
    for (int s = 0; s < 4; ++s) {
        const int ncol = n0 + s * 16 + (lane & 15);
        const float bias = pb[ncol];
#pragma unroll
        for (int r = 0; r < 8; ++r)
            tokens[(long)(mrow0 + r) * D_MODEL + ncol] = acc[s][r] + bias;
    }
}

// ---------------------------------------------------------------------------
// Fused residual-add + LayerNorm, one wave (32 lanes) per row of 192.
// mode 0: out = LN(in)            (resid unused)
// mode 1: resid = in;  out = LN(resid)
// mode 2: resid += in; out = LN(resid)
// ---------------------------------------------------------------------------
__global__ void ln_fused(const float* __restrict__ in, long in_stride,
                         float* __restrict__ resid,
                         const float* __restrict__ w, const float* __restrict__ bsh,
                         float* __restrict__ out, long out_stride,
                         int rows, int mode) {
    const int wave = threadIdx.x >> 5;
    const int lane = threadIdx.x & 31;
    const int row = blockIdx.x * (blockDim.x >> 5) + wave;
    if (row >= rows) return;

    const float* ip = in + (long)row * in_stride;
    float* rp = resid ? resid + (long)row * D_MODEL : nullptr;

    float v[6];
#pragma unroll
    for (int u = 0; u < 6; ++u) {
        const int e = lane + 32 * u;
        float x = ip[e];
        if (mode == 1)      { rp[e] = x; }
        else if (mode == 2) { x += rp[e]; rp[e] = x; }
        v[u] = x;
    }
    float s = 0.0f, s2 = 0.0f;
#pragma unroll
    for (int u = 0; u < 6; ++u) { s += v[u]; s2 += v[u] * v[u]; }
#pragma unroll
    for (int off = 16; off > 0; off >>= 1) {
        s  += __shfl_xor(s,  off, 32);
        s2 += __shfl_xor(s2, off, 32);
    }
    const float mu  = s * (1.0f / D_MODEL);
    const float var = s2 * (1.0f / D_MODEL) - mu * mu;
    const float inv = rsqrtf(var + 1e-5f);

    float* op = out + (long)row * out_stride;
#pragma unroll
    for (int u = 0; u < 6; ++u) {
        const int e = lane + 32 * u;
        op[e] = (v[u] - mu) * inv * w[e] + bsh[e];
    }
}

// ---------------------------------------------------------------------------
// Depthwise causal conv (K=4) + SiLU over x_in = xz[..., :D_INNER].
// ---------------------------------------------------------------------------
__global__ void conv_silu(const float* __restrict__ xz,
                          const float* __restrict__ cw,  // (D_INNER, 4)
                          const float* __restrict__ cb,  // (D_INNER)
                          float* __restrict__ xc) {
    const int idx = blockIdx.x * blockDim.x + threadIdx.x;
    if (idx >= ROWS_BL * D_INNER) return;
    const int d  = idx % D_INNER;
    const long bl = idx / D_INNER;            // b*196 + l
    const int l  = (int)(bl % LSEQ);
    const long base = bl - l;                 // b*196

    float acc = cb[d];
#pragma unroll
    for (int k = 0; k < D_CONV; ++k) {
        const int ls = l - (D_CONV - 1) + k;
        if (ls >= 0)
            acc += xz[(base + ls) * (2 * D_INNER) + d] * cw[d * D_CONV + k];
    }
    xc[idx] = acc * (1.0f / (1.0f + __expf(-acc)));
}

// ---------------------------------------------------------------------------
// Selective scan, fused with D*xc skip and SiLU(z) gating.
// One block per batch element, 384 threads (= channels), state in registers,
// per-step B/C (16 each) broadcast through LDS.
// ---------------------------------------------------------------------------
__global__ void scan_fused(const float* __restrict__ delta,   // (B*L, 384)
                           const float* __restrict__ xc,      // (B*L, 384)
                           const float* __restrict__ dbc,     // (B*L, 44)
                           const float* __restrict__ xz,      // (B*L, 768) -> z at +384
                           const float* __restrict__ A_log,   // (384, 16)
                           const float* __restrict__ Dv,      // (384)
                           float* __restrict__ y) {           // (B*L, 384)
    __shared__ float sB[D_STATE];
    __shared__ float sC[D_STATE];
    const int b = blockIdx.x;
    const int d = threadIdx.x;

    float Arow[D_STATE];
#pragma unroll
    for (int n = 0; n < D_STATE; ++n)
        Arow[n] = -__expf(A_log[d * D_STATE + n]);
    float h[D_STATE];
#pragma unroll
    for (int n = 0; n < D_STATE; ++n) h[n] = 0.0f;
    const float Dp = Dv[d];

    for (int l = 0; l < LSEQ; ++l) {
        const long rb = (long)b * LSEQ + l;
        if (d < D_STATE)            sB[d]           = dbc[rb * 44 + DT_RANK + d];
        else if (d < 2 * D_STATE)   sC[d - D_STATE] = dbc[rb * 44 + DT_RANK + D_STATE + (d - D_STATE)];
        __syncthreads();

        const float dl = delta[rb * D_INNER + d];
        const float xv = xc[rb * D_INNER + d];
        float yv = 0.0f;
#pragma unroll
        for (int n = 0; n < D_STATE; ++n) {
            h[n] = __expf(dl * Arow[n]) * h[n] + dl * sB[n] * xv;
            yv += h[n] * sC[n];
        }
        yv += Dp * xv;
        const float zv = xz[rb * (2 * D_INNER) + D_INNER + d];
        yv *= silu_f(zv);
        y[rb * D_INNER + d] = yv;
        __syncthreads();
    }
}

// ---------------------------------------------------------------------------
// Host orchestration
// ---------------------------------------------------------------------------
static inline void launch_gemm(const float* A, long lda, const float* W,
                               const float* bias, float* C, long ldc,
                               int M, int N, int K, int epi, hipStream_t s) {
    const int nblocks = (N + 63) / 64;
    const int total = (M / 16) * nblocks;
    const int wpb = 8;                            // 8 waves / 256 threads
    const int blocks = (total + wpb - 1) / wpb;
    gemm_wmma_f32<<<blocks, wpb * 32, 0, s>>>(A, lda, W, bias, C, ldc,
                                              M, N, K, nblocks, epi);
}

extern "C" void kernel_launch(void* const* d_in, const int* in_sizes, int n_in,
                              void* d_out, int out_size, void* d_ws, size_t ws_size,
                              hipStream_t stream) {
    const float* x         = (const float*)d_in[0];
    const float* patch_w   = (const float*)d_in[1];
    const float* patch_b   = (const float*)d_in[2];
    const float* norm_w    = (const float*)d_in[3];
    const float* norm_b    = (const float*)d_in[4];
    const float* in_proj_w = (const float*)d_in[5];
    const float* conv_w    = (const float*)d_in[6];
    const float* conv_b    = (const float*)d_in[7];
    const float* x_proj_w  = (const float*)d_in[8];
    const float* dt_proj_w = (const float*)d_in[9];
    const float* dt_proj_b = (const float*)d_in[10];
    const float* A_log     = (const float*)d_in[11];
    const float* Dvec      = (const float*)d_in[12];
    const float* out_proj_w= (const float*)d_in[13];
    const float* normf_w   = (const float*)d_in[14];
    const float* normf_b   = (const float*)d_in[15];
    const float* head_w    = (const float*)d_in[16];
    const float* head_b    = (const float*)d_in[17];
    float* out = (float*)d_out;

    // Workspace carve-up (all float counts even -> 8B alignment preserved).
    float* ws       = (float*)d_ws;
    float* hidden   = ws;                                  // 12544*192
    float* residual = hidden   + (long)ROWS_BL * D_MODEL;
    float* hn       = residual + (long)ROWS_BL * D_MODEL;
    float* xz       = hn       + (long)ROWS_BL * D_MODEL;  // 12544*768
    float* xc       = xz       + (long)ROWS_BL * 2 * D_INNER;
    float* dbc      = xc       + (long)ROWS_BL * D_INNER;  // 12544*44
    float* delta    = dbc      + (long)ROWS_BL * 44;
    float* ybuf     = delta    + (long)ROWS_BL * D_INNER;
    float* feat     = ybuf     + (long)ROWS_BL * D_INNER;  // 64*192

    // 1) patch embedding -> hidden (tokens)
    {
        const int total = (ROWS_BL / 16) * (D_MODEL / 64);
        const int wpb = 8;
        patch_embed_wmma<<<(total + wpb - 1) / wpb, wpb * 32, 0, stream>>>(
            x, patch_w, patch_b, hidden);
    }

    const int ln_blocks = (ROWS_BL + 7) / 8;   // 8 rows per 256-thread block

    for (int i = 0; i < DEPTH; ++i) {
        const float* nw  = norm_w    + (long)i * D_MODEL;
        const float* nb  = norm_b    + (long)i * D_MODEL;
        const float* ipw = in_proj_w + (long)i * 2 * D_INNER * D_MODEL;
        const float* cw  = conv_w    + (long)i * D_INNER * D_CONV;
        const float* cb  = conv_b    + (long)i * D_INNER;
        const float* xpw = x_proj_w  + (long)i * (DT_RANK + 2 * D_STATE) * D_INNER;
        const float* dtw = dt_proj_w + (long)i * D_INNER * DT_RANK;
        const float* dtb = dt_proj_b + (long)i * D_INNER;
        const float* Al  = A_log     + (long)i * D_INNER * D_STATE;
        const float* Dp  = Dvec      + (long)i * D_INNER;
        const float* opw = out_proj_w+ (long)i * D_MODEL * D_INNER;

        // residual update + LayerNorm -> hn
        ln_fused<<<ln_blocks, 256, 0, stream>>>(
            hidden, D_MODEL, residual, nw, nb, hn, D_MODEL,
            ROWS_BL, (i == 0) ? 1 : 2);

        // in_proj: xz = hn @ ipw^T    (12544 x 768, K=192)
        launch_gemm(hn, D_MODEL, ipw, nullptr, xz, 2 * D_INNER,
                    ROWS_BL, 2 * D_INNER, D_MODEL, 0, stream);

        // depthwise conv + SiLU -> xc
        {
            const int tot = ROWS_BL * D_INNER;
            conv_silu<<<(tot + 255) / 256, 256, 0, stream>>>(xz, cw, cb, xc);
        }

        // x_proj: dbc = xc @ xpw^T    (12544 x 44, K=384)
        launch_gemm(xc, D_INNER, xpw, nullptr, dbc, DT_RANK + 2 * D_STATE,
                    ROWS_BL, DT_RANK + 2 * D_STATE, D_INNER, 0, stream);

        // dt_proj + softplus: delta = softplus(dbc[:, :12] @ dtw^T + dtb)
        launch_gemm(dbc, DT_RANK + 2 * D_STATE, dtw, dtb, delta, D_INNER,
                    ROWS_BL, D_INNER, DT_RANK, 1, stream);

        // selective scan (+ D skip + SiLU(z) gate) -> ybuf
        scan_fused<<<BATCH, D_INNER, 0, stream>>>(
            delta, xc, dbc, xz, Al, Dp, ybuf);

        // out_proj: hidden = ybuf @ opw^T   (12544 x 192, K=384)
        launch_gemm(ybuf, D_INNER, opw, nullptr, hidden, D_MODEL,
                    ROWS_BL, D_MODEL, D_INNER, 0, stream);
    }

    // final LayerNorm on last token of each batch -> feat (64 x 192)
    ln_fused<<<(BATCH + 7) / 8, 256, 0, stream>>>(
        hidden + (long)(LSEQ - 1) * D_MODEL, (long)LSEQ * D_MODEL,
        nullptr, normf_w, normf_b, feat, D_MODEL, BATCH, 0);

    // head: out = feat @ head_w^T + head_b   (64 x 1000, K=192)
    launch_gemm(feat, D_MODEL, head_w, head_b, out, NCLS,
                BATCH, NCLS, D_MODEL, 0, stream);

    (void)in_sizes; (void)n_in; (void)out_size; (void)ws_size;
}